// BaseGraph_27951647163109
// MI455X (gfx1250) — compile-verified
//
#include <hip/hip_runtime.h>

#define NNODES 100000
#define NEDGES 1600000
#define FIN    128
#define FH     128
#define FC     40

typedef __attribute__((ext_vector_type(2))) float v2f;
typedef __attribute__((ext_vector_type(8))) float v8f;

// ---------------- degree / normalization ----------------

__global__ void __launch_bounds__(256) k_fill1(float* __restrict__ deg) {
    int i = blockIdx.x * 256 + threadIdx.x;
    if (i < NNODES) deg[i] = 1.0f;   // self-loop contributes 1
}

__global__ void __launch_bounds__(256) k_degacc(const int* __restrict__ col,
                                                float* __restrict__ deg) {
    int e = blockIdx.x * 256 + threadIdx.x;
    if (e < NEDGES) atomicAdd(&deg[col[e]], 1.0f);
}

__global__ void __launch_bounds__(256) k_rsqrt(float* __restrict__ deg) {
    int i = blockIdx.x * 256 + threadIdx.x;
    if (i < NNODES) deg[i] = rsqrtf(deg[i]);   // deg >= 1 always (self loops)
}

// ---------------- GEMM1: h1[N,128] = x[N,128] @ W1[128,128] (f32 WMMA) ----------------

__global__ void __launch_bounds__(256)
k_gemm1(const float* __restrict__ x, const float* __restrict__ W,
        float* __restrict__ h) {
    const int lane = threadIdx.x & 31;
    const int wid  = threadIdx.x >> 5;
    const int idx  = blockIdx.x * 8 + wid;     // 781*8 = 6248 tiles (of 50000)
    const int mt   = idx >> 3;                 // 0..780
    const int nt   = idx & 7;                  // 0..7
    const int half = lane >> 4;                // 0/1 lane group
    const int m    = lane & 15;

    const float* arow = x + (size_t)(mt * 16 + m) * FIN;
    const int    col  = nt * 16 + m;

    v8f acc = {};
#pragma unroll
    for (int k0 = 0; k0 < FIN; k0 += 4) {
        // A 16x4 f32: lane group holds K = k0+2*half .. +1 (contiguous -> v2f load)
        v2f a = *(const v2f*)(arow + k0 + 2 * half);
        // B 4x16 f32: VGPR v, group g -> row K = k0 + v + 2*g
        v2f b;
        b.x = W[(size_t)(k0 + 2 * half    ) * FH + col];
        b.y = W[(size_t)(k0 + 2 * half + 1) * FH + col];
        acc = __builtin_amdgcn_wmma_f32_16x16x4_f32(
            false, a, false, b, (short)0, acc, false, false);
    }
    // D 16x16 f32: VGPR v, group g -> row M = v + 8*g, col N = lane&15
    float* outp = h + (size_t)(mt * 16) * FH + nt * 16 + m;
#pragma unroll
    for (int v = 0; v < 8; ++v)
        outp[(size_t)(v + half * 8) * FH] = acc[v];
}

// ---------------- init agg1 = b1 + dinv^2 * h1 (self-loop + bias folded) ----------------

__global__ void __launch_bounds__(256)
k_self1(const float* __restrict__ h1, const float* __restrict__ dinv,
        const float* __restrict__ b1, float* __restrict__ agg1) {
    int tid = blockIdx.x * 256 + threadIdx.x;    // N*32 threads, float4 each
    int i = tid >> 5, q = tid & 31;
    float s = dinv[i]; s *= s;
    float4 hv = *(const float4*)(h1 + (size_t)i * FH + q * 4);
    float4 bv = *(const float4*)(b1 + q * 4);
    float4 o;
    o.x = bv.x + s * hv.x;  o.y = bv.y + s * hv.y;
    o.z = bv.z + s * hv.z;  o.w = bv.w + s * hv.w;
    *(float4*)(agg1 + (size_t)i * FH + q * 4) = o;
}

// ---------------- layer-1 edge scatter: agg1[t] += h1[s] * dinv[s]*dinv[t] ----------------

__global__ void __launch_bounds__(256)
k_scat1(const int* __restrict__ row, const int* __restrict__ col,
        const float* __restrict__ dinv, const float* __restrict__ h1,
        float* __restrict__ agg1) {
    int tid = blockIdx.x * 256 + threadIdx.x;    // E*32 threads (wave == one edge)
    int e = tid >> 5, q = tid & 31;
    int s = row[e], t = col[e];
    float w = dinv[s] * dinv[t];
    float4 hv = *(const float4*)(h1 + (size_t)s * FH + q * 4);
    float* dst = agg1 + (size_t)t * FH + q * 4;
    atomicAdd(dst + 0, hv.x * w);
    atomicAdd(dst + 1, hv.y * w);
    atomicAdd(dst + 2, hv.z * w);
    atomicAdd(dst + 3, hv.w * w);
}

// ---------------- GEMM2: g[N,40] = relu(agg1)[N,128] @ W2[128,40] (f32 WMMA) ----------------

__global__ void __launch_bounds__(256)
k_gemm2(const float* __restrict__ ag, const float* __restrict__ W2,
        float* __restrict__ g, int ntiles) {
    const int idx = blockIdx.x * 8 + (threadIdx.x >> 5);
    if (idx >= ntiles) return;                  // uniform per-wave exit (EXEC stays full)
    const int lane = threadIdx.x & 31;
    const int mt   = idx / 3;
    const int nt   = idx - mt * 3;              // 3 N-tiles cover 48 >= 40 cols
    const int half = lane >> 4;
    const int m    = lane & 15;
    const int col  = nt * 16 + m;
    const bool cok = (col < FC);

    const float* arow = ag + (size_t)(mt * 16 + m) * FH;

    v8f acc = {};
#pragma unroll
    for (int k0 = 0; k0 < FH; k0 += 4) {
        v2f av = *(const v2f*)(arow + k0 + 2 * half);
        v2f a;
        a.x = fmaxf(av.x, 0.0f);                // ReLU folded into A-fragment load
        a.y = fmaxf(av.y, 0.0f);
        v2f b;
        b.x = cok ? W2[(size_t)(k0 + 2 * half    ) * FC + col] : 0.0f;
        b.y = cok ? W2[(size_t)(k0 + 2 * half + 1) * FC + col] : 0.0f;
        acc = __builtin_amdgcn_wmma_f32_16x16x4_f32(
            false, a, false, b, (short)0, acc, false, false);
    }
    if (cok) {
        float* outp = g + (size_t)(mt * 16) * FC + col;
#pragma unroll
        for (int v = 0; v < 8; ++v)
            outp[(size_t)(v + half * 8) * FC] = acc[v];
    }
}

// ---------------- init out = b2 + dinv^2 * g ----------------

__global__ void __launch_bounds__(256)
k_self2(const float* __restrict__ g, const float* __restrict__ dinv,
        const float* __restrict__ b2, float* __restrict__ out) {
    int tid = blockIdx.x * 256 + threadIdx.x;    // N*40 threads
    int i = tid / FC;
    int f = tid - i * FC;
    float s = dinv[i]; s *= s;
    out[tid] = b2[f] + s * g[tid];
}

// ---------------- layer-2 edge scatter: out[t] += g[s] * dinv[s]*dinv[t] ----------------

__global__ void __launch_bounds__(256)
k_scat2(const int* __restrict__ row, const int* __restrict__ col,
        const float* __restrict__ dinv, const float* __restrict__ g,
        float* __restrict__ out) {
    int tid = blockIdx.x * 256 + threadIdx.x;    // E*10 threads, float4 each
    int e = tid / 10;
    int q = tid - e * 10;                        // 10 float4 = 40 floats
    int s = row[e], t = col[e];
    float w = dinv[s] * dinv[t];
    float4 gv = *(const float4*)(g + (size_t)s * FC + q * 4);
    float* dst = out + (size_t)t * FC + q * 4;
    atomicAdd(dst + 0, gv.x * w);
    atomicAdd(dst + 1, gv.y * w);
    atomicAdd(dst + 2, gv.z * w);
    atomicAdd(dst + 3, gv.w * w);
}

// ---------------- host side ----------------

extern "C" void kernel_launch(void* const* d_in, const int* in_sizes, int n_in,
                              void* d_out, int out_size, void* d_ws, size_t ws_size,
                              hipStream_t stream) {
    (void)in_sizes; (void)n_in; (void)out_size; (void)ws_size;

    const float* x   = (const float*)d_in[0];
    const int*   ei  = (const int*)d_in[1];
    const int*   row = ei;            // edge_index[0]: sources
    const int*   col = ei + NEDGES;   // edge_index[1]: targets
    const float* W1  = (const float*)d_in[2];
    const float* b1  = (const float*)d_in[3];
    const float* W2  = (const float*)d_in[4];
    const float* b2  = (const float*)d_in[5];
    float* out = (float*)d_out;

    char* ws = (char*)d_ws;
    float* dinv = (float*)ws;                                   // N floats (deg, then rsqrt)
    float* h1   = (float*)(ws + (size_t)(4u  << 20));           // N*128 f32 (51.2 MB)
    float* agg1 = (float*)(ws + (size_t)(60u << 20));           // N*128 f32 (51.2 MB)
    float* g    = h1;                                            // alias: h1 dead after scat1

    // normalization
    k_fill1 <<<(NNODES + 255) / 256, 256, 0, stream>>>(dinv);
    k_degacc<<<NEDGES / 256,         256, 0, stream>>>(col, dinv);
    k_rsqrt <<<(NNODES + 255) / 256, 256, 0, stream>>>(dinv);

    // layer 1
    k_gemm1 <<<NNODES / 16 / 8,      256, 0, stream>>>(x, W1, h1);          // 781 blocks
    k_self1 <<<NNODES * 32 / 256,    256, 0, stream>>>(h1, dinv, b1, agg1); // 12500
    k_scat1 <<<NEDGES * 32 / 256,    256, 0, stream>>>(row, col, dinv, h1, agg1); // 200000

    // layer 2
    const int ntiles2 = (NNODES / 16) * 3;                                   // 18750
    k_gemm2 <<<(ntiles2 + 7) / 8,    256, 0, stream>>>(agg1, W2, g, ntiles2);
    k_self2 <<<NNODES * FC / 256,    256, 0, stream>>>(g, dinv, b2, out);    // 15625
    k_scat2 <<<NEDGES * 10 / 256,    256, 0, stream>>>(row, col, dinv, g, out); // 62500
}